// GraphLayer_47785806135663
// MI455X (gfx1250) — compile-verified
//
#include <hip/hip_runtime.h>

// Problem constants (match reference)
#define NN   20000
#define EE   320000
#define DIN  128
#define HH1  256
#define HH2  256
#define DOUT 128

typedef float v2f __attribute__((ext_vector_type(2)));
typedef float v8f __attribute__((ext_vector_type(8)));

// ---------------- degree / normalization ----------------
__global__ void deg_init_kernel(float* __restrict__ deg, int n) {
    int i = blockIdx.x * blockDim.x + threadIdx.x;
    if (i < n) deg[i] = 1.0f;   // self-loop
}

__global__ void deg_acc_kernel(const int* __restrict__ dst, float* __restrict__ deg, int e) {
    int i = blockIdx.x * blockDim.x + threadIdx.x;
    if (i < e) unsafeAtomicAdd(&deg[dst[i]], 1.0f);
}

__global__ void dinv_kernel(const float* __restrict__ deg, float* __restrict__ dinv, int n) {
    int i = blockIdx.x * blockDim.x + threadIdx.x;
    if (i < n) dinv[i] = rsqrtf(deg[i]);
}

// ---------------- fp32 WMMA GEMM: H[n,FOUT] = act(X)[n,FIN] @ W[FIN,FOUT] ----------------
// One wave computes a 16(M) x 128(N) tile via 8 accumulators, stepping K by 4.
// When APPLY_ACT, PReLU is applied to the A operand on load (fuses previous
// layer's activation into this GEMM, saving a full N*F read+write sweep).
template <int FIN, int FOUT, bool APPLY_ACT>
__global__ __launch_bounds__(256) void gemm_wmma_f32(const float* __restrict__ X,
                                                     const float* __restrict__ W,
                                                     float* __restrict__ H,
                                                     const float* __restrict__ alpha,
                                                     int nrows) {
    constexpr int NGROUP = FOUT / 128;             // column groups of 128
    const int strips = nrows >> 4;                 // nrows multiple of 16
    const int wave = blockIdx.x * (blockDim.x >> 5) + (threadIdx.x >> 5);
    if (wave >= strips * NGROUP) return;
    const int s = wave / NGROUP;                   // 16-row strip
    const int g = wave % NGROUP;                   // 128-col group
    const int lane = threadIdx.x & 31;
    const int m16  = lane & 15;
    const int half = lane >> 4;

    const float al = APPLY_ACT ? alpha[0] : 0.0f;
    const float* xrow = X + (size_t)((s << 4) + m16) * FIN;

    v8f acc[8] = {};

    for (int k0 = 0; k0 < FIN; k0 += 4) {
        // A fragment: A[m][k0 + 2*half + {0,1}]  (contiguous pair -> b64 load)
        v2f a;
        a.x = xrow[k0 + 2 * half];
        a.y = xrow[k0 + 2 * half + 1];
        if (APPLY_ACT) {
            a.x = (a.x >= 0.0f) ? a.x : al * a.x;   // fused PReLU
            a.y = (a.y >= 0.0f) ? a.y : al * a.y;
        }
        const float* wr0 = W + (size_t)(k0 + 2 * half) * FOUT + g * 128 + m16;
        const float* wr1 = wr0 + FOUT;
#pragma unroll
        for (int t = 0; t < 8; ++t) {
            // B fragment: B[k0 + 2*half + vgpr][n]
            v2f b;
            b.x = wr0[t * 16];
            b.y = wr1[t * 16];
            acc[t] = __builtin_amdgcn_wmma_f32_16x16x4_f32(
                /*neg_a=*/false, a, /*neg_b=*/false, b,
                /*c_mod=*/(short)0, acc[t], /*reuse_a=*/false, /*reuse_b=*/false);
        }
    }

    // C/D layout: VGPR r, lanes 0-15 -> M=r, N=lane; lanes 16-31 -> M=r+8, N=lane-16
#pragma unroll
    for (int t = 0; t < 8; ++t) {
        const int col = g * 128 + t * 16 + m16;
#pragma unroll
        for (int r = 0; r < 8; ++r) {
            const int orow = (s << 4) + r + 8 * half;
            H[(size_t)orow * FOUT + col] = acc[t][r];
        }
    }
}

// ---------------- agg init: agg = h * norm_self + bias ----------------
__global__ void init_agg_kernel(const float* __restrict__ h,
                                const float* __restrict__ dinv,
                                const float* __restrict__ bias,
                                float* __restrict__ agg, int n, int f) {
    int idx = blockIdx.x * blockDim.x + threadIdx.x;
    const int fq = f >> 2;
    if (idx >= n * fq) return;
    const int node = idx / fq;
    const int fc   = (idx - node * fq) << 2;
    const float di = dinv[node];
    const float ns = di * di;
    const float4 hv = reinterpret_cast<const float4*>(h)[idx];
    float4 r;
    r.x = hv.x * ns + bias[fc + 0];
    r.y = hv.y * ns + bias[fc + 1];
    r.z = hv.z * ns + bias[fc + 2];
    r.w = hv.w * ns + bias[fc + 3];
    reinterpret_cast<float4*>(agg)[idx] = r;
}

// ---------------- edge scatter: agg[dst] += h[src] * dinv[src]*dinv[dst] ----------------
__global__ void edge_scatter_kernel(const float* __restrict__ h,
                                    const float* __restrict__ dinv,
                                    const int* __restrict__ src,
                                    const int* __restrict__ dst,
                                    float* __restrict__ agg, int e, int f) {
    int idx = blockIdx.x * blockDim.x + threadIdx.x;
    const int fq = f >> 2;
    if (idx >= e * fq) return;
    const int ei = idx / fq;
    const int fc = (idx - ei * fq) << 2;
    const int s = src[ei];
    const int d = dst[ei];
    const float ne = dinv[s] * dinv[d];
    const float4 hv = *reinterpret_cast<const float4*>(h + (size_t)s * f + fc);
    float* out = agg + (size_t)d * f + fc;
    unsafeAtomicAdd(out + 0, hv.x * ne);
    unsafeAtomicAdd(out + 1, hv.y * ne);
    unsafeAtomicAdd(out + 2, hv.z * ne);
    unsafeAtomicAdd(out + 3, hv.w * ne);
}

// ---------------- host-side orchestration ----------------
static inline int ceil_div(int a, int b) { return (a + b - 1) / b; }

extern "C" void kernel_launch(void* const* d_in, const int* in_sizes, int n_in,
                              void* d_out, int out_size, void* d_ws, size_t ws_size,
                              hipStream_t stream) {
    const float* X  = (const float*)d_in[0];
    const float* W1 = (const float*)d_in[1];
    const float* b1 = (const float*)d_in[2];
    const float* W2 = (const float*)d_in[3];
    const float* b2 = (const float*)d_in[4];
    const float* W3 = (const float*)d_in[5];
    const float* b3 = (const float*)d_in[6];
    const float* pa = (const float*)d_in[7];
    const int* eidx = (const int*)d_in[8];
    const int* src = eidx;
    const int* dst = eidx + EE;
    float* out = (float*)d_out;

    // workspace carve (floats): deg[N] | dinv[N] | hbuf[N*256] | aggbuf[N*256]
    float* ws     = (float*)d_ws;
    float* deg    = ws;
    float* dinv   = ws + NN;
    float* hbuf   = ws + 2 * NN;
    float* aggbuf = hbuf + (size_t)NN * 256;

    const int TPB = 256;

    // normalization (shared across batches)
    deg_init_kernel<<<ceil_div(NN, TPB), TPB, 0, stream>>>(deg, NN);
    deg_acc_kernel<<<ceil_div(EE, TPB), TPB, 0, stream>>>(dst, deg, EE);
    dinv_kernel<<<ceil_div(NN, TPB), TPB, 0, stream>>>(deg, dinv, NN);

    for (int b = 0; b < 2; ++b) {
        const float* Xb = X + (size_t)b * NN * DIN;
        float* outb     = out + (size_t)b * NN * DOUT;

        // ---- layer 1: DIN=128 -> H1=256 (no input activation) ----
        {
            const int waves = (NN / 16) * (HH1 / 128);
            gemm_wmma_f32<DIN, HH1, false><<<ceil_div(waves, 8), TPB, 0, stream>>>(Xb, W1, hbuf, pa, NN);
            init_agg_kernel<<<ceil_div(NN * (HH1 / 4), TPB), TPB, 0, stream>>>(hbuf, dinv, b1, aggbuf, NN, HH1);
            edge_scatter_kernel<<<ceil_div(EE * (HH1 / 4), TPB), TPB, 0, stream>>>(hbuf, dinv, src, dst, aggbuf, EE, HH1);
        }
        // ---- layer 2: H1=256 -> H2=256 (PReLU fused into GEMM A-load) ----
        {
            const int waves = (NN / 16) * (HH2 / 128);
            gemm_wmma_f32<HH1, HH2, true><<<ceil_div(waves, 8), TPB, 0, stream>>>(aggbuf, W2, hbuf, pa, NN);
            init_agg_kernel<<<ceil_div(NN * (HH2 / 4), TPB), TPB, 0, stream>>>(hbuf, dinv, b2, aggbuf, NN, HH2);
            edge_scatter_kernel<<<ceil_div(EE * (HH2 / 4), TPB), TPB, 0, stream>>>(hbuf, dinv, src, dst, aggbuf, EE, HH2);
        }
        // ---- layer 3: H2=256 -> DOUT=128 (PReLU fused; agg directly into d_out) ----
        {
            const int waves = (NN / 16) * (DOUT / 128);
            gemm_wmma_f32<HH2, DOUT, true><<<ceil_div(waves, 8), TPB, 0, stream>>>(aggbuf, W3, hbuf, pa, NN);
            init_agg_kernel<<<ceil_div(NN * (DOUT / 4), TPB), TPB, 0, stream>>>(hbuf, dinv, b3, outb, NN, DOUT);
            edge_scatter_kernel<<<ceil_div(EE * (DOUT / 4), TPB), TPB, 0, stream>>>(hbuf, dinv, src, dst, outb, EE, DOUT);
        }
    }
}